// MYIR2_63995012710464
// MI455X (gfx1250) — compile-verified
//
#include <hip/hip_runtime.h>

#ifndef __has_builtin
#define __has_builtin(x) 0
#endif

#define IAC_C 128
#define IAC_H 64
#define IAC_W 64
#define IAC_NPIX (IAC_H * IAC_W)   // 4096
#define IAC_NITER 17
#define IAC_KT 3
#define ROWS_PER_THREAD 16
#define THREADS 256

// ---- CDNA5 async global->LDS path (probe-guarded) ---------------------------
#if __has_builtin(__builtin_amdgcn_global_load_async_to_lds_b128)
#define HAVE_ASYNC_LDS 1
#else
#define HAVE_ASYNC_LDS 0
#endif
#if __has_builtin(__builtin_amdgcn_s_wait_asynccnt)
#define HAVE_WAIT_ASYNC 1
#else
#define HAVE_WAIT_ASYNC 0
#endif

typedef int iac_v4i __attribute__((ext_vector_type(4)));
#if HAVE_ASYNC_LDS
typedef __attribute__((address_space(1))) iac_v4i* iac_g4p;  // global int4*
typedef __attribute__((address_space(3))) iac_v4i* iac_l4p;  // LDS int4*
#endif

__device__ __forceinline__ void async_copy_b128(float* lds_dst, const float* gsrc) {
#if defined(__HIP_DEVICE_COMPILE__) && HAVE_ASYNC_LDS
  // GLOBAL_LOAD_ASYNC_TO_LDS_B128: tracked with ASYNCcnt, no VGPR data path
  __builtin_amdgcn_global_load_async_to_lds_b128(
      (iac_g4p)gsrc, (iac_l4p)lds_dst, /*offset=*/0, /*cpol=*/0);
#else
  *(float4*)lds_dst = *(const float4*)gsrc;  // fallback: load + ds_store
#endif
}

template <int N>
__device__ __forceinline__ void wait_async() {
#if defined(__HIP_DEVICE_COMPILE__) && HAVE_ASYNC_LDS
#if HAVE_WAIT_ASYNC
  __builtin_amdgcn_s_wait_asynccnt((unsigned short)N);
#else
  asm volatile("s_wait_asynccnt %0" ::"i"(N) : "memory");
#endif
#endif
}

// Each thread stages 16 floats (4 x B128) of one 64x64 bias plane into LDS.
__device__ __forceinline__ void issue_bias(float* dst_plane, const float* src_plane, int t) {
  const int base = t * 16;
#pragma unroll
  for (int j = 0; j < 4; ++j)
    async_copy_b128(dst_plane + base + 4 * j, src_plane + base + 4 * j);
}

// ---- main kernel: one workgroup per channel --------------------------------
__global__ __launch_bounds__(THREADS) void iac_sac17_kernel(
    const float* __restrict__ feat_in,
    const float* __restrict__ kern_F,
    float* __restrict__ out) {
  __shared__ __align__(16) float f[IAC_NPIX];        // feature plane (resident)
  __shared__ __align__(16) float fv[IAC_NPIX];       // vertical-pass intermediate
  __shared__ __align__(16) float sbias[2][IAC_NPIX]; // async double-buffered bias

  const int ch = blockIdx.x;
  const int t  = threadIdx.x;
  const int x  = t & (IAC_W - 1);            // lane owns column x ...
  const int y0 = (t >> 6) * ROWS_PER_THREAD; // ... rows y0..y0+15
  const int xm = (x == 0) ? 0 : x - 1;       // replicate-pad clamps (x const per thread)
  const int xp = (x == IAC_W - 1) ? x : x + 1;

  // Load input feature plane (row-contiguous float4, fully coalesced).
  {
    const float4* src = (const float4*)(feat_in + (size_t)ch * IAC_NPIX);
    float4* dst = (float4*)f;
#pragma unroll
    for (int j = 0; j < IAC_NPIX / 4 / THREADS; ++j)
      dst[t + j * THREADS] = src[t + j * THREADS];
  }

  // kern_F layout: iter i taps at channels i*2*C*K + ch*3 + {0,1,2};
  //               iter i bias at channel  N*2*C*K + i*C + ch.
  const float* bias0 =
      kern_F + ((size_t)(IAC_NITER * 2 * IAC_C * IAC_KT) + (size_t)ch) * IAC_NPIX;
  issue_bias(sbias[0], bias0, t);  // prime the async pipeline for iter 0

  for (int i = 0; i < IAC_NITER; ++i) {
    __syncthreads();  // prev horizontal done everywhere: f valid, sbias[(i+1)&1] free

    if (i + 1 < IAC_NITER)
      issue_bias(sbias[(i + 1) & 1],
                 bias0 + (size_t)(i + 1) * IAC_C * IAC_NPIX, t);

    const float* tb =
        kern_F + ((size_t)i * (2 * IAC_C * IAC_KT) + (size_t)ch * IAC_KT) * IAC_NPIX;

    // Warm L2 for next iteration's 48KB tap block (192B per thread).
    if (i + 1 < IAC_NITER) {
      const char* ntb = (const char*)(tb + (size_t)(2 * IAC_C * IAC_KT) * IAC_NPIX);
      __builtin_prefetch(ntb + (size_t)t * 192, 0, 1);
      __builtin_prefetch(ntb + (size_t)t * 192 + 128, 0, 1);
    }

    // Per-pixel taps -> registers, used by both passes. Coalesced 128B/wave.
    float tp0[ROWS_PER_THREAD], tp1[ROWS_PER_THREAD], tp2[ROWS_PER_THREAD];
#pragma unroll
    for (int r = 0; r < ROWS_PER_THREAD; ++r) {
      const size_t p = (size_t)(y0 + r) * IAC_W + x;
      tp0[r] = tb[p];
      tp1[r] = tb[IAC_NPIX + p];
      tp2[r] = tb[2 * IAC_NPIX + p];
    }

    // Vertical pass: 18 LDS reads per thread (rows y0-1 .. y0+16, edge-clamped).
    float col[ROWS_PER_THREAD + 2];
#pragma unroll
    for (int r = 0; r < ROWS_PER_THREAD + 2; ++r) {
      int y = y0 + r - 1;
      y = (y < 0) ? 0 : ((y > IAC_H - 1) ? IAC_H - 1 : y);
      col[r] = f[y * IAC_W + x];
    }
#pragma unroll
    for (int r = 0; r < ROWS_PER_THREAD; ++r)
      fv[(y0 + r) * IAC_W + x] =
          fmaf(tp0[r], col[r], fmaf(tp1[r], col[r + 1], tp2[r] * col[r + 2]));

    // Ensure this iteration's async bias landed (in-order ASYNCcnt), then sync
    // so all waves see fv and every wave's share of the bias plane.
    if (i + 1 < IAC_NITER) wait_async<4>(); else wait_async<0>();
    __syncthreads();

    // Horizontal pass + bias + leaky_relu(0.1); writes f for next iteration.
    const float* bi = sbias[i & 1];
#pragma unroll
    for (int r = 0; r < ROWS_PER_THREAD; ++r) {
      const int row = (y0 + r) * IAC_W;
      const float a = fv[row + xm];
      const float b = fv[row + x];
      const float c = fv[row + xp];
      float v = fmaf(tp0[r], a, fmaf(tp1[r], b, tp2[r] * c)) + bi[row + x];
      v = (v >= 0.0f) ? v : 0.1f * v;
      f[row + x] = v;
    }
  }

  __syncthreads();
  {
    float4* dst = (float4*)(out + (size_t)ch * IAC_NPIX);
    const float4* src = (const float4*)f;
#pragma unroll
    for (int j = 0; j < IAC_NPIX / 4 / THREADS; ++j)
      dst[t + j * THREADS] = src[t + j * THREADS];
  }
}

extern "C" void kernel_launch(void* const* d_in, const int* in_sizes, int n_in,
                              void* d_out, int out_size, void* d_ws, size_t ws_size,
                              hipStream_t stream) {
  (void)in_sizes; (void)n_in; (void)out_size; (void)d_ws; (void)ws_size;
  const float* feat = (const float*)d_in[0];
  const float* kern = (const float*)d_in[1];
  float* out = (float*)d_out;
  iac_sac17_kernel<<<dim3(IAC_C), dim3(THREADS), 0, stream>>>(feat, kern, out);
}